// LIFCell_74775380623691
// MI455X (gfx1250) — compile-verified
//
#include <hip/hip_runtime.h>
#include <cstdint>
#include <cstddef>

// LIF scan: v = v*d*(1-z) + x_t ; z = (v > 0.5) ? 1 : 0
// x_seq (B,T,H) f32 -> spk (B,T,H) f32.  Pure streaming problem:
// 512 MiB @ 23.3 TB/s ~= 22 us floor, ~0.4 GFLOP total -> bandwidth-bound.
// CDNA5 path: deep ASYNCcnt-tracked global->LDS prefetch ring + NT cache policy.
// (No WMMA: the op contains no matmul.)

#define STAGES 32              // async prefetch depth (ASYNCcnt is 6-bit, 32 <= 63)
#define BLOCK  64              // threads per block = 2 wave32s
#define VEC    4               // floats per thread (b128 transfers)
#define HTILE  (BLOCK * VEC)   // 256 h-values per block

typedef float v4f __attribute__((ext_vector_type(4)));

__global__ __launch_bounds__(BLOCK) void lif_scan_async(
    const float* __restrict__ x,
    const float* __restrict__ decay,
    float* __restrict__ out,
    int T, int H)
{
    // Per-lane-private ring buffer in LDS: STAGES slots of BLOCK 16B entries.
    __shared__ v4f smem[STAGES * BLOCK];

    const int tid = threadIdx.x;
    const int ht  = blockIdx.x;          // H tile index   (grid.x = H/HTILE)
    const int b   = blockIdx.y;          // batch index    (grid.y = B)
    const int h0  = ht * HTILE + tid * VEC;

    // d = sigmoid(decay[h0..h0+3]) once per thread
    v4f dec = *reinterpret_cast<const v4f*>(decay + h0);
    v4f d;
    d.x = 1.0f / (1.0f + __expf(-dec.x));
    d.y = 1.0f / (1.0f + __expf(-dec.y));
    d.z = 1.0f / (1.0f + __expf(-dec.z));
    d.w = 1.0f / (1.0f + __expf(-dec.w));

    const uint32_t strideB  = (uint32_t)H * 4u;  // bytes per timestep
    const uint32_t base_off = ((uint32_t)b * (uint32_t)T * (uint32_t)H + (uint32_t)h0) * 4u;

    // Low 32 bits of the flat shared address == LDS byte offset (ISA aperture rule).
    const uint32_t lds_lane = (uint32_t)(uintptr_t)(&smem[tid]);
    const uint32_t slot_b   = (uint32_t)(BLOCK * sizeof(v4f));   // 1024 B per stage

    uint32_t issue_off = base_off;   // global byte offset of next timestep to prefetch

    // ---- Prologue: fill the ring with STAGES async loads (ASYNCcnt += STAGES) ----
#pragma unroll
    for (int s = 0; s < STAGES; ++s) {
        uint32_t laddr = lds_lane + (uint32_t)s * slot_b;
        asm volatile("global_load_async_to_lds_b128 %0, %1, %2 th:TH_LOAD_NT"
                     :: "v"(laddr), "v"(issue_off), "s"(x)
                     : "memory");
        issue_off += strideB;
    }

    v4f v = (v4f)0.0f;
    v4f z = (v4f)0.0f;
    uint32_t out_off = base_off;
    int slot = 0;

#pragma unroll 4
    for (int t = 0; t < T; ++t) {
        // Oldest outstanding async load (slot `slot`) has landed in LDS.
        asm volatile("s_wait_asynccnt %0" :: "i"(STAGES - 1) : "memory");

        v4f xv = smem[slot * BLOCK + tid];             // ds_load_b128

        // WAR guard: DS read must retire before the DMA overwrites this slot.
        asm volatile("s_wait_dscnt 0" ::: "memory");

        if (t + STAGES < T) {
            uint32_t laddr = lds_lane + (uint32_t)slot * slot_b;
            asm volatile("global_load_async_to_lds_b128 %0, %1, %2 th:TH_LOAD_NT"
                         :: "v"(laddr), "v"(issue_off), "s"(x)
                         : "memory");
            issue_off += strideB;
        }

        // LIF update (z in {0,1})
        v.x = v.x * d.x * (1.0f - z.x) + xv.x;
        v.y = v.y * d.y * (1.0f - z.y) + xv.y;
        v.z = v.z * d.z * (1.0f - z.z) + xv.z;
        v.w = v.w * d.w * (1.0f - z.w) + xv.w;
        z.x = (v.x > 0.5f) ? 1.0f : 0.0f;
        z.y = (v.y > 0.5f) ? 1.0f : 0.0f;
        z.z = (v.z > 0.5f) ? 1.0f : 0.0f;
        z.w = (v.w > 0.5f) ? 1.0f : 0.0f;

        // Streaming store, non-temporal (write-once data; keep L2 clean).
        __builtin_nontemporal_store(
            z, reinterpret_cast<v4f*>(reinterpret_cast<char*>(out) + out_off));
        out_off += strideB;

        slot = (slot + 1) & (STAGES - 1);
    }
}

extern "C" void kernel_launch(void* const* d_in, const int* in_sizes, int n_in,
                              void* d_out, int out_size, void* d_ws, size_t ws_size,
                              hipStream_t stream) {
    const float* x     = (const float*)d_in[0];   // (B, T, H) f32
    const float* decay = (const float*)d_in[1];   // (H,) f32
    float* out         = (float*)d_out;           // (B, T, H) f32

    const int H = in_sizes[1];          // 1024
    const int T = 512;                  // fixed by reference
    const int B = in_sizes[0] / (T * H);

    dim3 grid(H / HTILE, B);            // (4, 128) = 512 blocks x 64 threads = 1024 wave32s
    lif_scan_async<<<grid, BLOCK, 0, stream>>>(x, decay, out, T, H);
}